// MIND_46909632807094
// MI455X (gfx1250) — compile-verified
//
#include <hip/hip_runtime.h>

// MIND multi-interest forward for MI455X (gfx1250, wave32).
// One workgroup (8 wave32) per batch row. Gather-bound problem (~160MB/call
// -> ~7us at 23.3 TB/s HBM). The dense hat GEMM (100x64x64 f32 per row) runs
// on V_WMMA_F32_16X16X4_F32 (full f32 precision); the hist_emb gather uses
// the CDNA5 one-hop async path GLOBAL_LOAD_ASYNC_TO_LDS_B128 (ASYNCcnt).

#define L_SZ 100
#define K_SZ 4
#define NNEG_SZ 200
#define HS 68          // padded LDS row stride (floats): 272B = 16B aligned, conflict-free
#define LPAD 112       // 100 rounded up to 7*16 for WMMA M tiles

typedef __attribute__((ext_vector_type(2))) float v2f;
typedef __attribute__((ext_vector_type(8))) float v8f;

__device__ __forceinline__ float wave_sum(float v) {
#pragma unroll
  for (int m = 16; m >= 1; m >>= 1) v += __shfl_xor(v, m, 32);
  return v;
}

__device__ __forceinline__ float wave_max(float v) {
#pragma unroll
  for (int m = 16; m >= 1; m >>= 1) v = fmaxf(v, __shfl_xor(v, m, 32));
  return v;
}

// Cross-lane LDS RAW inside one wave: wait out DScnt explicitly.
__device__ __forceinline__ void lds_fence() {
  asm volatile("s_wait_dscnt 0" ::: "memory");
}

// One-hop async copy: 16B per lane, global -> LDS, tracked by ASYNCcnt.
// lds_off is the per-lane LDS byte offset (low 32 bits of the flat shared
// address per the LDS aperture mapping); gaddr is the 64-bit global address.
__device__ __forceinline__ void async_gather_b128(unsigned lds_off,
                                                  unsigned long long gaddr) {
  asm volatile("global_load_async_to_lds_b128 %0, %1, off"
               :: "v"(lds_off), "v"(gaddr) : "memory");
}
__device__ __forceinline__ void async_fence() {
  asm volatile("s_wait_asynccnt 0" ::: "memory");
}

__global__ __launch_bounds__(256) void mind_fwd(
    const int* __restrict__ user_id, const int* __restrict__ hist,
    const int* __restrict__ pos_item, const int* __restrict__ neg_items,
    const float* __restrict__ routing_init, const float* __restrict__ user_table,
    const float* __restrict__ item_table, const float* __restrict__ W_cap,
    const float* __restrict__ W_convert, float* __restrict__ out)
{
  __shared__ __align__(16) float s_hist[LPAD * HS];  // hist_emb; reused later
  __shared__ __align__(16) float s_hat [LPAD * HS];  // hat = hist_emb @ W_cap

  // Reuse of s_hist after the WMMA barrier (disjoint float offsets):
  float* s_cap = s_hist;           // [4][64]
  float* s_ue  = s_hist + 256;     // [4][64]
  float* s_pos = s_hist + 512;     // [64]
  float* s_neg = s_hist + 576;     // [4][64]
  float* s_sw  = s_hist + 832;     // [4][100]

  const int b    = blockIdx.x;
  const int tid  = threadIdx.x;
  const int lane = tid & 31;
  const int wave = tid >> 5;

  // ------- Phase 1: async gather hist_emb -> LDS (16B/lane, ASYNCcnt) -----
  const int* hb = hist + (long long)b * L_SZ;
  for (int i = tid; i < L_SZ * 16; i += 256) {
    const int row = i >> 4;
    const int c   = (i & 15) << 2;
    const long long idx = hb[row];
    const unsigned lds_off =
        (unsigned)(unsigned long long)(uintptr_t)&s_hist[row * HS + c];
    async_gather_b128(lds_off,
                      (unsigned long long)(uintptr_t)(item_table + idx * 64 + c));
  }
  for (int i = tid; i < 12 * 17; i += 256) {  // zero rows 100..111 (68 floats each)
    const int row = L_SZ + i / 17;
    const int c   = (i % 17) << 2;
    *(float4*)&s_hist[row * HS + c] = make_float4(0.f, 0.f, 0.f, 0.f);
  }
  async_fence();   // this wave's async copies landed in LDS
  __syncthreads(); // all waves' copies visible

  // ---------------- Phase 2: hat = hist_emb @ W_cap via f32 WMMA -----------
  {
    const int half = lane >> 4;   // K half-select per ISA A/B f32 layout
    const int lt   = lane & 15;
    for (int t = wave; t < 28; t += 8) {       // 7 M-tiles x 4 N-tiles
      const int m0 = (t >> 2) * 16;
      const int n0 = (t & 3) * 16;
      const int arow = (m0 + lt) * HS;
      v8f acc = {};
#pragma unroll
      for (int kk = 0; kk < 64; kk += 4) {
        const int k = kk + half * 2;
        v2f A, Bf;
        A.x  = s_hist[arow + k];
        A.y  = s_hist[arow + k + 1];
        Bf.x = W_cap[k * 64 + n0 + lt];
        Bf.y = W_cap[(k + 1) * 64 + n0 + lt];
        acc = __builtin_amdgcn_wmma_f32_16x16x4_f32(false, A, false, Bf,
                                                    (short)0, acc, false, false);
      }
#pragma unroll
      for (int r = 0; r < 8; ++r) {
        const int row = m0 + r + half * 8;     // C/D layout: VGPR r -> M=r / r+8
        s_hat[row * HS + n0 + lt] = acc[r];
      }
    }
  }
  __syncthreads();  // hat complete; s_hist region now free for reuse

  if (wave < K_SZ) {
    // ------------- Phase 3a (waves 0-3): routing for interest k ------------
    const int k  = wave;
    const int d0 = lane * 2;
    float wreg[4];
    bool  valid[4], mk[4];
#pragma unroll
    for (int j = 0; j < 4; ++j) {
      const int l = j * 32 + lane;
      valid[j] = (l < L_SZ);
      if (valid[j]) {
        wreg[j] = routing_init[((long long)b * K_SZ + k) * L_SZ + l];
        mk[j]   = hb[l] > 0;
      } else { wreg[j] = 0.f; mk[j] = false; }
    }

    float c0 = 0.f, c1 = 0.f;
    for (int it = 0; it < 3; ++it) {
      // softmax over full L (mask applied afterwards, as in reference)
      float mx = -3.4e38f;
#pragma unroll
      for (int j = 0; j < 4; ++j) if (valid[j]) mx = fmaxf(mx, wreg[j]);
      mx = wave_max(mx);
      float e[4]; float ssum = 0.f;
#pragma unroll
      for (int j = 0; j < 4; ++j) {
        e[j] = valid[j] ? __expf(wreg[j] - mx) : 0.f;
        ssum += e[j];
      }
      ssum = wave_sum(ssum);
      const float rinv = 1.f / ssum;
#pragma unroll
      for (int j = 0; j < 4; ++j) {
        const int l = j * 32 + lane;
        if (valid[j]) s_sw[k * L_SZ + l] = mk[j] ? e[j] * rinv : 0.f;
      }
      lds_fence();

      // cap[d] = sum_l sw[l] * hat[l][d]  (lane owns dims d0, d0+1)
      c0 = 0.f; c1 = 0.f;
      for (int l = 0; l < L_SZ; ++l) {
        const float s  = s_sw[k * L_SZ + l];               // uniform -> broadcast
        const float2 h = *(const float2*)&s_hat[l * HS + d0];
        c0 = fmaf(s, h.x, c0);
        c1 = fmaf(s, h.y, c1);
      }
      // squash
      const float n  = wave_sum(c0 * c0 + c1 * c1);
      const float sc = n / (1.f + n) / sqrtf(n + 1e-9f);
      c0 *= sc; c1 *= sc;
      s_cap[k * 64 + d0]     = c0;
      s_cap[k * 64 + d0 + 1] = c1;
      lds_fence();

      if (it < 2) {  // w[l] += dot(hat[l,:], cap)
#pragma unroll
        for (int j = 0; j < 4; ++j) {
          const int l = j * 32 + lane;
          if (l < L_SZ) {
            float dot = 0.f;
            for (int dd = 0; dd < 64; dd += 4) {
              const float4 h = *(const float4*)&s_hat[l * HS + dd];
              const float4 c = *(const float4*)&s_cap[k * 64 + dd];
              dot += h.x * c.x + h.y * c.y + h.z * c.z + h.w * c.w;
            }
            wreg[j] += dot;
          }
        }
      }
    }

    // ------------- user tower: ue[k] = l2norm([u ; cap_k] @ W_convert) -----
    const long long uid = user_id[b];
    const float* u = user_table + uid * 64;
    float a0 = 0.f, a1 = 0.f;
    for (int j = 0; j < 64; ++j) {
      const float uj = u[j];                    // uniform -> scalar load
      const float cj = s_cap[k * 64 + j];       // uniform -> LDS broadcast
      const float2 w0 = *(const float2*)(W_convert + j * 64 + d0);
      const float2 w1 = *(const float2*)(W_convert + (64 + j) * 64 + d0);
      a0 += uj * w0.x + cj * w1.x;
      a1 += uj * w0.y + cj * w1.y;
    }
    const float nn  = wave_sum(a0 * a0 + a1 * a1);
    const float inv = 1.f / fmaxf(sqrtf(nn), 1e-12f);
    s_ue[k * 64 + d0]     = a0 * inv;
    s_ue[k * 64 + d0 + 1] = a1 * inv;
  } else {
    // ------------- Phase 3b (waves 4-7): item tower (dominant gathers) -----
    const int wv = wave - 4;
    const int d0 = lane * 2;
    const int* nb = neg_items + (long long)b * NNEG_SZ;
    float a0 = 0.f, a1 = 0.f;
    for (int n = wv; n < NNEG_SZ; n += 4) {
      const long long idx = nb[n];
      const float2 v  = *(const float2*)(item_table + idx * 64 + d0);
      const float ss  = wave_sum(v.x * v.x + v.y * v.y);
      const float inv = 1.f / fmaxf(sqrtf(ss), 1e-12f);
      a0 = fmaf(v.x, inv, a0);
      a1 = fmaf(v.y, inv, a1);
    }
    s_neg[wv * 64 + d0]     = a0;
    s_neg[wv * 64 + d0 + 1] = a1;
    if (wv == 0) {
      const long long pidx = pos_item[b];
      const float2 v  = *(const float2*)(item_table + pidx * 64 + d0);
      const float ss  = wave_sum(v.x * v.x + v.y * v.y);
      const float inv = 1.f / fmaxf(sqrtf(ss), 1e-12f);
      s_pos[d0]     = v.x * inv;
      s_pos[d0 + 1] = v.y * inv;
    }
  }
  __syncthreads();

  // ---------------- Phase 4: interest selection + output (wave 0) ----------
  if (wave == 0) {
    const int d0 = lane * 2;
    const float p0 = s_pos[d0], p1 = s_pos[d0 + 1];
    float sum0 = p0, sum1 = p1;                 // pos + sum of 200 normalized negs
#pragma unroll
    for (int wv = 0; wv < 4; ++wv) {
      sum0 += s_neg[wv * 64 + d0];
      sum1 += s_neg[wv * 64 + d0 + 1];
    }
    float best = -3.4e38f;
    int bk = 0;
#pragma unroll
    for (int kk = 0; kk < K_SZ; ++kk) {         // first-max = jnp.argmax tie-break
      const float dot =
          wave_sum(s_ue[kk * 64 + d0] * p0 + s_ue[kk * 64 + d0 + 1] * p1);
      if (dot > best) { best = dot; bk = kk; }
    }
    out[(long long)b * 64 + d0]     = s_ue[bk * 64 + d0]     * sum0;
    out[(long long)b * 64 + d0 + 1] = s_ue[bk * 64 + d0 + 1] * sum1;
  }
}

extern "C" void kernel_launch(void* const* d_in, const int* in_sizes, int n_in,
                              void* d_out, int out_size, void* d_ws, size_t ws_size,
                              hipStream_t stream) {
  const int*   user_id      = (const int*)d_in[0];
  const int*   hist         = (const int*)d_in[1];
  const int*   pos_item     = (const int*)d_in[2];
  const int*   neg_items    = (const int*)d_in[3];
  const float* routing_init = (const float*)d_in[4];
  const float* user_table   = (const float*)d_in[5];
  const float* item_table   = (const float*)d_in[6];
  const float* W_cap        = (const float*)d_in[7];
  const float* W_convert    = (const float*)d_in[8];
  float* out = (float*)d_out;

  const int B = in_sizes[0];  // 2048
  mind_fwd<<<dim3(B), dim3(256), 0, stream>>>(
      user_id, hist, pos_item, neg_items, routing_init,
      user_table, item_table, W_cap, W_convert, out);
}